// TGNNPO_83837761618033
// MI455X (gfx1250) — compile-verified
//
#include <hip/hip_runtime.h>
#include <math.h>

typedef __attribute__((ext_vector_type(2))) float v2f;
typedef __attribute__((ext_vector_type(8))) float v8f;

#define N_NODES 50000
#define F_DIM   128
#define P_DIM   12
#define MID_DIM 3
#define E_EDGES 800000
#define HID     (P_DIM * MID_DIM)   // 36
#define NTILES  (N_NODES / 16)      // 3125
#define WAVES_PER_BLOCK 8
#define KPAIRS  (F_DIM / 2)         // 64 float2 K-pairs per period

// ---------------------------------------------------------------------------
// 1) Degree: deg[dst] += 1 per edge
// ---------------------------------------------------------------------------
__global__ void tg_deg_kernel(const long long* __restrict__ dst,
                              float* __restrict__ deg, int E) {
    int e = blockIdx.x * blockDim.x + threadIdx.x;
    if (e < E) atomicAdd(&deg[(int)dst[e]], 1.0f);
}

// ---------------------------------------------------------------------------
// 2) Fused dual-projection GEMM, all 12 periods per wave, 2 periods per C.
//    - One wave owns one 16-node tile.
//    - Period 2j uses B columns 0..5 (Wl|Wr, zeros elsewhere); period 2j+1
//      uses columns 6..11. Both accumulate into cacc[j] -> 6 x v8f = 48 VGPRs.
//    - A: each lane loads x[node][kA..kA+1][0..11] = 24 contiguous floats
//      (6 x b128): every cacheline of x is read exactly once.
//    - B: staged in LDS as float2 K-pairs; 12 batched ds_load_b64 per k-step.
// ---------------------------------------------------------------------------
__global__ void __launch_bounds__(256, 1)
tg_proj_wmma_kernel(const float* __restrict__ x,
                    const float* __restrict__ Wl,
                    const float* __restrict__ Wr,
                    const float* __restrict__ bl,
                    float* __restrict__ proj,
                    float* __restrict__ xr) {
    // bsh[p][q][col] = { B_p[2q][col], B_p[2q+1][col] }; even p occupies
    // cols 0..5, odd p occupies cols 6..11, zeros elsewhere. 96 KB.
    __shared__ v2f bsh[P_DIM * KPAIRS * 16];

    // ---- cooperative B staging (12288 float2, 48 per thread) ----
    for (int idx = threadIdx.x; idx < P_DIM * KPAIRS * 16; idx += blockDim.x) {
        const int col = idx & 15;
        const int q   = (idx >> 4) & (KPAIRS - 1);
        const int p   = idx >> 10;                  // idx / (64*16)
        const int k   = q * 2;
        const int c   = col - ((p & 1) ? 6 : 0);    // 0..5 -> active slot
        float b0 = 0.f, b1 = 0.f;
        if (c >= 0 && c < 2 * MID_DIM) {
            const int  m = (c < MID_DIM) ? c : (c - MID_DIM);
            const float* w = ((c < MID_DIM) ? Wl : Wr) + p * F_DIM * MID_DIM;
            b0 = w[k * MID_DIM + m];
            b1 = w[(k + 1) * MID_DIM + m];
        }
        v2f t; t.x = b0; t.y = b1;
        bsh[idx] = t;
    }
    __syncthreads();

    const int wid  = threadIdx.x >> 5;
    const int tile = blockIdx.x * WAVES_PER_BLOCK + wid;
    if (tile >= NTILES) return;                      // wave-uniform exit

    const int lane = threadIdx.x & 31;
    const int half = lane >> 4;                      // K-half (A & B rows)
    const int col  = lane & 15;                      // A row / B,C column

    // A base: lane covers features kA = k0 + 2*half, all 12 periods
    const float* xbase = x + (size_t)(tile * 16 + col) * (F_DIM * P_DIM)
                           + (size_t)(2 * half) * P_DIM;

    v8f cacc[P_DIM / 2];
    #pragma unroll
    for (int j = 0; j < P_DIM / 2; ++j)
        cacc[j] = (v8f){0.f, 0.f, 0.f, 0.f, 0.f, 0.f, 0.f, 0.f};

    #pragma unroll 1
    for (int k0 = 0; k0 < F_DIM; k0 += 4) {
        // 24 contiguous floats: x[node][kA][0..11] ++ x[node][kA+1][0..11]
        const float4* xp4 =
            reinterpret_cast<const float4*>(xbase + (size_t)k0 * P_DIM);
        float abuf[24];
        #pragma unroll
        for (int i = 0; i < 6; ++i) {
            const float4 t = xp4[i];
            abuf[4 * i + 0] = t.x; abuf[4 * i + 1] = t.y;
            abuf[4 * i + 2] = t.z; abuf[4 * i + 3] = t.w;
        }

        // Batch all 12 B pairs (one dscnt wait), then 12 WMMAs.
        const int q = (k0 >> 1) + half;              // (k0 + 2*half) / 2
        v2f bv[P_DIM];
        #pragma unroll
        for (int p = 0; p < P_DIM; ++p)
            bv[p] = bsh[(p * KPAIRS + q) * 16 + col];

        #pragma unroll
        for (int j = 0; j < P_DIM / 2; ++j) {        // even periods -> cacc[j]
            const int p = 2 * j;
            v2f a; a.x = abuf[p]; a.y = abuf[12 + p];
            cacc[j] = __builtin_amdgcn_wmma_f32_16x16x4_f32(
                          false, a, false, bv[p], (short)0, cacc[j],
                          false, false);
        }
        #pragma unroll
        for (int j = 0; j < P_DIM / 2; ++j) {        // odd periods -> cacc[j]
            const int p = 2 * j + 1;
            v2f a; a.x = abuf[p]; a.y = abuf[12 + p];
            cacc[j] = __builtin_amdgcn_wmma_f32_16x16x4_f32(
                          false, a, false, bv[p], (short)0, cacc[j],
                          false, false);
        }
    }

    // C layout: VGPR r holds row M = r + 8*half, column = col.
    // cols 0..2: proj[p=2j], 3..5: xr[p=2j], 6..8: proj[p=2j+1], 9..11: xr.
    if (col < 12) {
        const int  pb   = (col >= 6) ? 1 : 0;
        const int  cc   = col - 6 * pb;              // 0..5
        const bool isXr = (cc >= MID_DIM);
        const int  m    = isXr ? (cc - MID_DIM) : cc;
        #pragma unroll
        for (int j = 0; j < P_DIM / 2; ++j) {
            const int p = 2 * j + pb;
            #pragma unroll
            for (int r = 0; r < 8; ++r) {
                const int node = tile * 16 + r + 8 * half;
                const size_t off = (size_t)node * HID + p * MID_DIM + m;
                const float v = cacc[j][r];
                if (isXr) xr[off] = v + bl[p * MID_DIM + m];
                else      proj[off] = v;
            }
        }
    }
}

// ---------------------------------------------------------------------------
// 3) Edge scatter: agg[dst] += proj[src]  (one thread per edge*period)
// ---------------------------------------------------------------------------
__global__ void tg_scatter_kernel(const long long* __restrict__ src,
                                  const long long* __restrict__ dst,
                                  const float* __restrict__ proj,
                                  float* __restrict__ agg) {
    int tid = blockIdx.x * blockDim.x + threadIdx.x;
    if (tid >= E_EDGES * P_DIM) return;
    const int e = tid / P_DIM;
    const int p = tid % P_DIM;
    const int s = (int)src[e];
    const int d = (int)dst[e];
    const size_t so   = (size_t)s * HID + p * MID_DIM;
    const size_t doff = (size_t)d * HID + p * MID_DIM;
    #pragma unroll
    for (int m = 0; m < MID_DIM; ++m)
        atomicAdd(&agg[doff + m], proj[so + m]);
}

// ---------------------------------------------------------------------------
// 4) Finalize: mean-agg + xr, row-normalize, scale, leaky-relu, W_lin dot
// ---------------------------------------------------------------------------
__global__ void tg_finalize_kernel(const float* __restrict__ agg,
                                   const float* __restrict__ xr,
                                   const float* __restrict__ deg,
                                   const float* __restrict__ W_lin,
                                   const float* __restrict__ b_lin,
                                   const float* __restrict__ gumbel,
                                   float* __restrict__ logit) {
    int n = blockIdx.x * blockDim.x + threadIdx.x;
    if (n >= N_NODES) return;
    const float inv = 1.0f / fmaxf(deg[n], 1.0f);
    float acc = 0.0f;
    #pragma unroll
    for (int p = 0; p < P_DIM; ++p) {
        float o[MID_DIM];
        float sq = 0.0f;
        #pragma unroll
        for (int m = 0; m < MID_DIM; ++m) {
            const size_t idx = (size_t)n * HID + p * MID_DIM + m;
            o[m] = agg[idx] * inv + xr[idx];
            sq += o[m] * o[m];
        }
        const float nrm = fmaxf(sqrtf(sq), 1e-12f);
        const float sc = ((float)(p + 1) / (float)P_DIM) / nrm;
        #pragma unroll
        for (int m = 0; m < MID_DIM; ++m) {
            float h = o[m] * sc;
            h = (h > 0.0f) ? h : 0.01f * h;          // leaky_relu(0.01)
            acc += h * W_lin[p * MID_DIM + m];
        }
    }
    logit[n] = acc + b_lin[0] + gumbel[n];
}

// ---------------------------------------------------------------------------
// 5) Softmax over axis 0: single-block max + sum-of-exp reduction
// ---------------------------------------------------------------------------
__global__ void tg_softmax_reduce(const float* __restrict__ logit,
                                  float* __restrict__ red, int n) {
    __shared__ float sdata[1024];
    const int tid = threadIdx.x;

    float m = -INFINITY;
    for (int i = tid; i < n; i += blockDim.x) m = fmaxf(m, logit[i]);
    sdata[tid] = m;
    __syncthreads();
    for (int off = blockDim.x >> 1; off > 0; off >>= 1) {
        if (tid < off) sdata[tid] = fmaxf(sdata[tid], sdata[tid + off]);
        __syncthreads();
    }
    const float mx = sdata[0];
    __syncthreads();

    float s = 0.0f;
    for (int i = tid; i < n; i += blockDim.x) s += __expf(logit[i] - mx);
    sdata[tid] = s;
    __syncthreads();
    for (int off = blockDim.x >> 1; off > 0; off >>= 1) {
        if (tid < off) sdata[tid] += sdata[tid + off];
        __syncthreads();
    }
    if (tid == 0) { red[0] = mx; red[1] = sdata[0]; }
}

__global__ void tg_softmax_write(const float* __restrict__ logit,
                                 const float* __restrict__ red,
                                 float* __restrict__ out, int n) {
    int i = blockIdx.x * blockDim.x + threadIdx.x;
    if (i < n) out[i] = __expf(logit[i] - red[0]) / red[1];
}

// ---------------------------------------------------------------------------
extern "C" void kernel_launch(void* const* d_in, const int* in_sizes, int n_in,
                              void* d_out, int out_size, void* d_ws, size_t ws_size,
                              hipStream_t stream) {
    const float*     x      = (const float*)d_in[0];
    const long long* eidx   = (const long long*)d_in[1];   // (2, E) int64
    // d_in[2] = edge_weight: unused by the reference
    const float*     Wl     = (const float*)d_in[3];
    const float*     bl     = (const float*)d_in[4];
    const float*     Wr     = (const float*)d_in[5];
    const float*     W_lin  = (const float*)d_in[6];
    const float*     b_lin  = (const float*)d_in[7];
    const float*     gumbel = (const float*)d_in[8];
    float*           out    = (float*)d_out;

    const long long* src = eidx;
    const long long* dst = eidx + E_EDGES;

    // Workspace layout (floats): deg | agg | proj | xr | logit | red
    float* ws    = (float*)d_ws;
    float* deg   = ws;                                     // N
    float* agg   = deg  + N_NODES;                         // N*36
    float* proj  = agg  + (size_t)N_NODES * HID;           // N*36
    float* xr    = proj + (size_t)N_NODES * HID;           // N*36
    float* logit = xr   + (size_t)N_NODES * HID;           // N
    float* red   = logit + N_NODES;                        // 2

    // Zero deg + agg (contiguous)
    hipMemsetAsync(deg, 0, sizeof(float) * (size_t)N_NODES * (1 + HID), stream);

    // 1) degree
    tg_deg_kernel<<<(E_EDGES + 255) / 256, 256, 0, stream>>>(dst, deg, E_EDGES);

    // 2) fused dual GEMM (WMMA f32): one wave per 16-node tile, 12 periods
    const int gemmBlocks = (NTILES + WAVES_PER_BLOCK - 1) / WAVES_PER_BLOCK;
    tg_proj_wmma_kernel<<<gemmBlocks, WAVES_PER_BLOCK * 32, 0, stream>>>(
        x, Wl, Wr, bl, proj, xr);

    // 3) scatter-add over edges
    const long long scatterThreads = (long long)E_EDGES * P_DIM;
    tg_scatter_kernel<<<(int)((scatterThreads + 255) / 256), 256, 0, stream>>>(
        src, dst, proj, agg);

    // 4) finalize -> logits
    tg_finalize_kernel<<<(N_NODES + 255) / 256, 256, 0, stream>>>(
        agg, xr, deg, W_lin, b_lin, gumbel, logit);

    // 5) softmax over all N
    tg_softmax_reduce<<<1, 1024, 0, stream>>>(logit, red, N_NODES);
    tg_softmax_write<<<(N_NODES + 255) / 256, 256, 0, stream>>>(
        logit, red, out, N_NODES);
}